// BrainInspiredEmotionGraph_23656679866466
// MI455X (gfx1250) — compile-verified
//
#include <hip/hip_runtime.h>

// ---------------------------------------------------------------------------
// RGCN (17 nodes, 8 relations, D=H=2048, fp32) for MI455X / gfx1250.
// Memory-bound: streams W1/W2/root once; matrix math via v_wmma_f32_16x16x4_f32.
// ---------------------------------------------------------------------------

#define NNODES 17
#define NPAD   32          // padded node rows for 2x 16-row WMMA tiles
#define NSLOT  9           // 8 relations + root slot
#define DDIM   2048

typedef __attribute__((ext_vector_type(2))) float v2f;
typedef __attribute__((ext_vector_type(8))) float v8f;

// ---------------------------------------------------------------------------
// prep: build transposed, per-relation-aggregated activations
//   xT[slot][k][node], node-stride NPAD.  slot<8: mean over edges of type
//   `slot` into `node` of x[src];  slot==8: x itself (root term).
//   Rows 17..31 zero-padded.  x[m] = (m < nsig) ? xsig[m] : xmain[m].
// ---------------------------------------------------------------------------
__global__ __launch_bounds__(256) void prep_kernel(
    const float* __restrict__ xmain, const float* __restrict__ xsig, int nsig,
    const int* __restrict__ ei, const int* __restrict__ et, int E,
    float* __restrict__ xT)
{
  int slot = blockIdx.x / NPAD;
  int n    = blockIdx.x % NPAD;

  for (int j = 0; j < DDIM / 256; ++j) {
    int d = threadIdx.x + j * 256;
    float v = 0.0f;
    if (n < NNODES) {
      if (slot == 8) {
        v = (n < nsig) ? xsig[(size_t)n * DDIM + d]
                       : xmain[(size_t)n * DDIM + d];
      } else {
        float acc = 0.0f;
        int   cnt = 0;
        for (int e = 0; e < E; ++e) {
          if (et[e] == slot && ei[E + e] == n) {   // edge_index[1][e] == dst
            int m = ei[e];                          // edge_index[0][e] == src
            acc += (m < nsig) ? xsig[(size_t)m * DDIM + d]
                              : xmain[(size_t)m * DDIM + d];
            ++cnt;
          }
        }
        v = cnt ? acc / (float)cnt : 0.0f;
      }
    }
    xT[((size_t)slot * DDIM + d) * NPAD + n] = v;
  }
}

// ---------------------------------------------------------------------------
// gemm: one wave per (slot, 16-column strip).  Streams W_slot (or root)
// exactly once; A-fragments come from the tiny resident xT (L2-hot).
//   partial[slot][row 0..31][h] = (xT_slot)^T-tile @ W-tile
// WMMA f32 16x16x4 layouts (wave32):
//   A (16x4): lane L<16 holds M=L, VGPR0=K0, VGPR1=K1; lanes 16-31: K2/K3.
//   B (4x16): lane L<16 holds N=L row K0/K1; lanes 16-31 row K2/K3.
// ---------------------------------------------------------------------------
__global__ __launch_bounds__(128) void gemm_kernel(
    const float* __restrict__ xT,     // [NSLOT][DDIM][NPAD]
    const float* __restrict__ W,      // [8][DDIM][DDIM]
    const float* __restrict__ root,   // [DDIM][DDIM]
    float* __restrict__ partial)      // [NSLOT][NPAD][DDIM]
{
  const int wave = threadIdx.x >> 5;
  const int lane = threadIdx.x & 31;
  const int task = blockIdx.x * 4 + wave;     // 0 .. 9*128-1
  const int slot  = task >> 7;
  const int strip = task & 127;
  const int h0    = strip * 16;

  const int half = lane >> 4;                 // 0: K0/K1 pair, 1: K2/K3 pair
  const int hl   = lane & 15;

  const float* __restrict__ Wr =
      (slot < 8) ? (W + (size_t)slot * DDIM * DDIM) : root;
  const float* __restrict__ Xs = xT + (size_t)slot * DDIM * NPAD;

  // per-lane streaming pointers
  const float* wp = Wr + (size_t)(2 * half) * DDIM + h0 + hl;  // B frag
  const float* ap = Xs + (size_t)(2 * half) * NPAD + hl;       // A frag (M 0..15)

  v8f c0 = {};   // rows 0..15
  v8f c1 = {};   // rows 16..31 (only row 16 non-zero; padding handles M=17..31)

#pragma unroll 4
  for (int k0 = 0; k0 < DDIM; k0 += 4) {
    v2f b, a0, a1;
    b.x  = wp[0];
    b.y  = wp[DDIM];
    a0.x = ap[0];
    a0.y = ap[NPAD];
    a1.x = ap[16];
    a1.y = ap[NPAD + 16];
    c0 = __builtin_amdgcn_wmma_f32_16x16x4_f32(false, a0, false, b,
                                               (short)0, c0, false, false);
    c1 = __builtin_amdgcn_wmma_f32_16x16x4_f32(false, a1, false, b,
                                               (short)0, c1, false, false);
    wp += (size_t)4 * DDIM;
    ap += 4 * NPAD;
  }

  // C/D layout: VGPR i = row (i + 8*half) at column h0+hl
  size_t base = (size_t)slot * NPAD * DDIM + h0 + hl;
#pragma unroll
  for (int i = 0; i < 8; ++i) {
    partial[base + (size_t)(i + 8 * half) * DDIM]      = c0[i];
    partial[base + (size_t)(16 + i + 8 * half) * DDIM] = c1[i];
  }
}

// ---------------------------------------------------------------------------
// reduce: out[n][h] = (relu?)( sum_slot partial[slot][n][h] + bias[h] )
// ---------------------------------------------------------------------------
__global__ __launch_bounds__(256) void reduce_kernel(
    const float* __restrict__ partial, const float* __restrict__ bias,
    float* __restrict__ out, int relu)
{
  int idx = blockIdx.x * blockDim.x + threadIdx.x;
  if (idx >= NNODES * DDIM) return;
  int n = idx / DDIM;
  int h = idx % DDIM;
  float s = bias[h];
#pragma unroll
  for (int slot = 0; slot < NSLOT; ++slot)
    s += partial[((size_t)slot * NPAD + n) * DDIM + h];
  if (relu) s = fmaxf(s, 0.0f);
  out[idx] = s;
}

// ---------------------------------------------------------------------------
extern "C" void kernel_launch(void* const* d_in, const int* in_sizes, int n_in,
                              void* d_out, int out_size, void* d_ws, size_t ws_size,
                              hipStream_t stream) {
  const float* node_emb = (const float*)d_in[0];
  const float* sig      = (const float*)d_in[1];
  const float* W1       = (const float*)d_in[2];
  const float* root1    = (const float*)d_in[3];
  const float* bias1    = (const float*)d_in[4];
  const float* W2       = (const float*)d_in[5];
  const float* root2    = (const float*)d_in[6];
  const float* bias2    = (const float*)d_in[7];
  const int*   ei       = (const int*)d_in[8];
  const int*   et       = (const int*)d_in[9];
  const int    E        = in_sizes[9];          // number of (doubled) edges

  float* xT      = (float*)d_ws;                          // NSLOT*DDIM*NPAD
  float* partial = xT + (size_t)NSLOT * DDIM * NPAD;      // NSLOT*NPAD*DDIM
  float* h1      = partial + (size_t)NSLOT * NPAD * DDIM; // NNODES*DDIM

  const int prepGrid   = NSLOT * NPAD;                    // 288 blocks
  const int gemmGrid   = (NSLOT * 128) / 4;               // 288 blocks x 4 waves
  const int reduceGrid = (NNODES * DDIM + 255) / 256;

  // ---- layer 1 ----
  prep_kernel<<<prepGrid, 256, 0, stream>>>(node_emb, sig, 6, ei, et, E, xT);
  gemm_kernel<<<gemmGrid, 128, 0, stream>>>(xT, W1, root1, partial);
  reduce_kernel<<<reduceGrid, 256, 0, stream>>>(partial, bias1, h1, 1);

  // ---- layer 2 ----
  prep_kernel<<<prepGrid, 256, 0, stream>>>(h1, nullptr, 0, ei, et, E, xT);
  gemm_kernel<<<gemmGrid, 128, 0, stream>>>(xT, W2, root2, partial);
  reduce_kernel<<<reduceGrid, 256, 0, stream>>>(partial, bias2, (float*)d_out, 0);
}